// LSTMForSOC_67164698575369
// MI455X (gfx1250) — compile-verified
//
#include <hip/hip_runtime.h>

// ---------------------------------------------------------------------------
// Fused LSTM forward for MI455X (gfx1250, wave32).
//   B=256, T=4096, XC=8, SCC=4, IN=12, H=16, 4H=64.
// One wave32 workgroup owns a 16-row batch tile; weights + states live in
// VGPRs; all GEMMs use V_WMMA_F32_16X16X4_F32 (exact fp32).
// Activations: v_exp_f32 (native exp2, constants pre-folded) + v_rcp_f32.
// ---------------------------------------------------------------------------

typedef __attribute__((ext_vector_type(2))) float v2f;
typedef __attribute__((ext_vector_type(8))) float v8f;

#define LSTM_B   256
#define LSTM_T   4096
#define LSTM_XC  8
#define LSTM_IN  12
#define LSTM_H   16

#define LOG2E_F      1.4426950408889634f
#define NEG_LOG2E_F  (-1.4426950408889634f)
#define NEG_2LOG2E_F (-2.8853900817779268f)

__device__ __forceinline__ v2f ld2(const float* p) {
    return *reinterpret_cast<const v2f*>(p);   // 8B-aligned by construction
}
__device__ __forceinline__ v2f mk2(float a, float b) {
    v2f r; r.x = a; r.y = b; return r;
}
__device__ __forceinline__ v8f wmma4(v2f a, v2f b, v8f c) {
    // D = A(16x4,f32) * B(4x16,f32) + C(16x16,f32)
    return __builtin_amdgcn_wmma_f32_16x16x4_f32(false, a, false, b,
                                                 (short)0, c, false, false);
}
// fast 1/x: single v_rcp_f32 (~1 ulp)
__device__ __forceinline__ float frcp(float x) {
    return __builtin_amdgcn_rcpf(x);
}
// native exp2: single v_exp_f32
__device__ __forceinline__ float fexp2(float x) {
    return __builtin_amdgcn_exp2f(x);
}
// sigmoid(x) = 1 / (1 + 2^(-x*log2e)) : mul, exp, add, rcp
__device__ __forceinline__ float sigm(float x) {
    return frcp(1.0f + fexp2(x * NEG_LOG2E_F));
}
// tanh(x) = (1-u)/(1+u), u = 2^(-2x*log2e) : mul, exp, add, add, rcp, mul
__device__ __forceinline__ float tanh_fast(float x) {
    float u = fexp2(x * NEG_2LOG2E_F);
    return (1.0f - u) * frcp(1.0f + u);
}

__global__ __launch_bounds__(32)
void lstm_fused_kernel(const float* __restrict__ X,     // [B,T,XC]
                       const float* __restrict__ SC,    // [B,SCC]
                       const float* __restrict__ W_ih,  // [64,12]
                       const float* __restrict__ W_hh,  // [64,16]
                       const float* __restrict__ b_ih,  // [64]
                       const float* __restrict__ b_hh,  // [64]
                       const float* __restrict__ W_out, // [1,16]
                       const float* __restrict__ b_out, // [1]
                       float* __restrict__ out)         // [B,T]
{
    __shared__ __align__(16) float hbuf[LSTM_H * LSTM_H]; // 16x16 h transpose stage

    const int lane = threadIdx.x;        // 0..31
    const int col  = lane & 15;          // N (B-frag) / M (A-frag) index
    const int half = lane >> 4;          // 0|1
    const int kb   = half * 2;           // K sub-offset within a K=4 chunk
    const int b0   = blockIdx.x * 16;    // batch tile base

    // ---- one-time register preload of all weight fragments ----------------
    v2f bx[4][3];    // W_ih B-fragments: [n-tile][K-chunk]
    v2f bh[4][4];    // W_hh B-fragments
    v2f bo[4];       // W_out packed into column N==0 of a B-fragment
    float bias[4];
#pragma unroll
    for (int n = 0; n < 4; ++n) {
        const int row = n * 16 + col;                 // gate index 0..63
        bias[n] = b_ih[row] + b_hh[row];
#pragma unroll
        for (int q = 0; q < 3; ++q)
            bx[n][q] = ld2(W_ih + row * LSTM_IN + q * 4 + kb);
#pragma unroll
        for (int q = 0; q < 4; ++q)
            bh[n][q] = ld2(W_hh + row * LSTM_H + q * 4 + kb);
    }
#pragma unroll
    for (int q = 0; q < 4; ++q)
        bo[q] = (col == 0) ? mk2(W_out[q * 4 + kb], W_out[q * 4 + kb + 1])
                           : mk2(0.0f, 0.0f);
    const float ob = b_out[0];

    // static-feature A-fragment (K-chunk 2 of the input projection, k=8..11)
    const v2f a_sc = ld2(SC + (b0 + col) * 4 + kb);

    // ---- recurrent state --------------------------------------------------
    v2f ah[4];                            // h in A-fragment layout (K chunks)
#pragma unroll
    for (int q = 0; q < 4; ++q) ah[q] = mk2(0.0f, 0.0f);
    v8f cst;                              // cell state, C/D layout
#pragma unroll
    for (int j = 0; j < 8; ++j) cst[j] = 0.0f;

    const float* xrow = X + (size_t)(b0 + col) * LSTM_T * LSTM_XC;

    // prime x fragments for t=0
    v2f ax0 = ld2(xrow + kb);
    v2f ax1 = ld2(xrow + 4 + kb);

    for (int t = 0; t < LSTM_T; ++t) {
        // prefetch x fragments for t+1 (independent of the recurrence)
        v2f nx0 = ax0, nx1 = ax1;
        if (t + 1 < LSTM_T) {
            const float* p = xrow + (size_t)(t + 1) * LSTM_XC;
            nx0 = ld2(p + kb);
            nx1 = ld2(p + 4 + kb);
        }

        // gates[16,64] = bias + inp@W_ih.T + h@W_hh.T   (4 N-tiles of 16x16)
        // x-proj WMMAs first (ready early), recurrence WMMAs last so only 4
        // matrix ops sit on the h(t-1) -> h(t) dependent path.
        v8f acc[4];
#pragma unroll
        for (int n = 0; n < 4; ++n) {
            v8f c;
#pragma unroll
            for (int j = 0; j < 8; ++j) c[j] = bias[n];
            c = wmma4(ax0,  bx[n][0], c);      // input proj K=0..3
            c = wmma4(ax1,  bx[n][1], c);      // input proj K=4..7
            c = wmma4(a_sc, bx[n][2], c);      // static  proj K=8..11
            c = wmma4(ah[0], bh[n][0], c);     // recurrence K=0..3
            c = wmma4(ah[1], bh[n][1], c);
            c = wmma4(ah[2], bh[n][2], c);
            c = wmma4(ah[3], bh[n][3], c);
            acc[n] = c;
        }

        // activations + state update (torch gate order i,f,g,o), C/D layout
        v8f hnew;
#pragma unroll
        for (int j = 0; j < 8; ++j) {
            const float ig = sigm(acc[0][j]);
            const float fg = sigm(acc[1][j]);
            const float gg = tanh_fast(acc[2][j]);
            const float og = sigm(acc[3][j]);
            const float cc = fg * cst[j] + ig * gg;
            cst[j]  = cc;
            hnew[j] = og * tanh_fast(cc);
        }

        // D-layout h -> LDS -> A-layout fragments for next step & output head
#pragma unroll
        for (int j = 0; j < 8; ++j)
            hbuf[(j + half * 8) * LSTM_H + col] = hnew[j];
        __syncthreads();
#pragma unroll
        for (int q = 0; q < 4; ++q)
            ah[q] = ld2(&hbuf[col * LSTM_H + q * 4 + kb]);
        __syncthreads();

        // output head: out[16,1] = h[16,16] @ W_out.T + b_out (column 0 of D)
        v8f oacc;
#pragma unroll
        for (int j = 0; j < 8; ++j) oacc[j] = ob;
#pragma unroll
        for (int q = 0; q < 4; ++q)
            oacc = wmma4(ah[q], bo[q], oacc);
        if (col == 0) {
#pragma unroll
            for (int j = 0; j < 8; ++j)
                out[(size_t)(b0 + j + half * 8) * LSTM_T + t] = oacc[j];
        }

        ax0 = nx0; ax1 = nx1;
    }
}

extern "C" void kernel_launch(void* const* d_in, const int* in_sizes, int n_in,
                              void* d_out, int out_size, void* d_ws, size_t ws_size,
                              hipStream_t stream) {
    (void)in_sizes; (void)n_in; (void)d_ws; (void)ws_size; (void)out_size;
    const float* X     = (const float*)d_in[0];
    const float* SC    = (const float*)d_in[1];
    const float* W_ih  = (const float*)d_in[2];
    const float* W_hh  = (const float*)d_in[3];
    const float* b_ih  = (const float*)d_in[4];
    const float* b_hh  = (const float*)d_in[5];
    const float* W_out = (const float*)d_in[6];
    const float* b_out = (const float*)d_in[7];
    float* out = (float*)d_out;

    lstm_fused_kernel<<<dim3(LSTM_B / 16), dim3(32), 0, stream>>>(
        X, SC, W_ih, W_hh, b_ih, b_hh, W_out, b_out, out);
}